// _SparseIFFT_19490561590131
// MI455X (gfx1250) — compile-verified
//
#include <hip/hip_runtime.h>

typedef __attribute__((ext_vector_type(2))) float v2f;
typedef __attribute__((ext_vector_type(8))) float v8f;

#define NFFT 512
#define SUB  64
#define ROWS 16
#define STR  517      // LDS row stride (pad): conflict-free for stride-8 A reads
#define NT   256      // 8 wave32s per block

// in-register complex butterfly: (er[A],ei[A]) += w*(er[B],ei[B]); B gets the difference
#define BFLY(A, B, WR, WI) {                                  \
    const float _tr = (WR) * er[B] - (WI) * ei[B];            \
    const float _ti = (WR) * ei[B] + (WI) * er[B];            \
    er[B] = er[A] - _tr;  ei[B] = ei[A] - _ti;                \
    er[A] += _tr;         ei[A] += _ti; }

__global__ __launch_bounds__(NT)
void sparse_ifft_wmma(const float* __restrict__ re,  const float* __restrict__ im,
                      const float* __restrict__ Mre, const float* __restrict__ Mim,
                      const float* __restrict__ tw0r, const float* __restrict__ tw0i,
                      const float* __restrict__ tw1r, const float* __restrict__ tw1i,
                      const float* __restrict__ tw2r, const float* __restrict__ tw2i,
                      float* __restrict__ outr, float* __restrict__ outi)
{
    __shared__ float lr[ROWS][STR];
    __shared__ float li[ROWS][STR];
    __shared__ float twLr[448];   // stage0 @0 (64), stage1 @64 (128), stage2 @192 (256)
    __shared__ float twLi[448];

    const int tid    = threadIdx.x;
    const int wv     = tid >> 5;          // wave32
    const int lane   = tid & 31;
    const int lanelo = lane & 15;
    const int hi     = lane >> 4;
    const long rowBase = (long)blockIdx.x * ROWS;

    // ---- Phase 1: stage raw rows, b128 coalesced global -> LDS; twiddles -> LDS ----
    for (int idx = tid; idx < ROWS * NFFT / 4; idx += NT) {
        const int r  = idx >> 7;              // 128 float4 per row
        const int c4 = (idx & 127) << 2;
        const long g = (rowBase + r) * NFFT + c4;
        const float4 vr = *reinterpret_cast<const float4*>(re + g);
        const float4 vi = *reinterpret_cast<const float4*>(im + g);
        lr[r][c4+0] = vr.x; lr[r][c4+1] = vr.y; lr[r][c4+2] = vr.z; lr[r][c4+3] = vr.w;
        li[r][c4+0] = vi.x; li[r][c4+1] = vi.y; li[r][c4+2] = vi.z; li[r][c4+3] = vi.w;
    }
    for (int idx = tid; idx < 448; idx += NT) {
        float a, b;
        if (idx < 64)       { a = tw0r[idx];       b = tw0i[idx]; }
        else if (idx < 192) { a = tw1r[idx - 64];  b = tw1i[idx - 64]; }
        else                { a = tw2r[idx - 192]; b = tw2i[idx - 192]; }
        twLr[idx] = a; twLi[idx] = b;
    }
    __syncthreads();

    // ---- Phase 2: eight 64-pt complex sub-IDFTs via V_WMMA_F32_16X16X4_F32 ----
    // wave -> (ntile = 16 output cols, 4 chunks). Bit-reversal permutation folded
    // into A addressing: chunk c element k = raw column k*8 + bitrev3(c).
    const int ntile = wv & 3;
    const int cbase = (wv >> 2) << 2;
    const int ncol  = ntile * 16 + lanelo;
    const int selk  = hi * 2;             // fragment K select for high half-wave

    v2f Bre[16], Bim[16], Bni[16];        // B fragments: M_re, M_im, -M_im
#pragma unroll
    for (int ks = 0; ks < 16; ++ks) {
        const int k0 = ks * 4 + selk;
        Bre[ks].x = Mre[(k0 + 0) * SUB + ncol];
        Bre[ks].y = Mre[(k0 + 1) * SUB + ncol];
        Bim[ks].x = Mim[(k0 + 0) * SUB + ncol];
        Bim[ks].y = Mim[(k0 + 1) * SUB + ncol];
        Bni[ks].x = -Bim[ks].x;
        Bni[ks].y = -Bim[ks].y;
    }

    v8f accR[4], accI[4];
#pragma unroll
    for (int cc = 0; cc < 4; ++cc) {
        const int chunk = cbase + cc;
        const int brc = ((chunk & 1) << 2) | (chunk & 2) | ((chunk >> 2) & 1); // bitrev3
        v8f aR = {0.f,0.f,0.f,0.f,0.f,0.f,0.f,0.f};
        v8f aI = {0.f,0.f,0.f,0.f,0.f,0.f,0.f,0.f};
#pragma unroll
        for (int ks = 0; ks < 16; ++ks) {
            const int k0 = ks * 4 + selk;
            v2f Ar, Ai;
            Ar.x = lr[lanelo][(k0 + 0) * 8 + brc];
            Ar.y = lr[lanelo][(k0 + 1) * 8 + brc];
            Ai.x = li[lanelo][(k0 + 0) * 8 + brc];
            Ai.y = li[lanelo][(k0 + 1) * 8 + brc];
            // acc_re += Ar*Mre - Ai*Mim ; acc_im += Ar*Mim + Ai*Mre
            aR = __builtin_amdgcn_wmma_f32_16x16x4_f32(false, Ar, false, Bre[ks], (short)0, aR, false, false);
            aR = __builtin_amdgcn_wmma_f32_16x16x4_f32(false, Ai, false, Bni[ks], (short)0, aR, false, false);
            aI = __builtin_amdgcn_wmma_f32_16x16x4_f32(false, Ar, false, Bim[ks], (short)0, aI, false, false);
            aI = __builtin_amdgcn_wmma_f32_16x16x4_f32(false, Ai, false, Bre[ks], (short)0, aI, false, false);
        }
        accR[cc] = aR; accI[cc] = aI;
    }

    __syncthreads();  // all A reads complete -> safe to overwrite LDS in chunk order

    // C/D layout: VGPR v, lanes 0-15 -> (M=v, N=lane), lanes 16-31 -> (M=8+v, N=lane-16)
#pragma unroll
    for (int cc = 0; cc < 4; ++cc) {
        const int chunk = cbase + cc;
        const int col = chunk * SUB + ncol;
        const int r0  = hi * 8;
#pragma unroll
        for (int v = 0; v < 8; ++v) {
            lr[r0 + v][col] = accR[cc][v];
            li[r0 + v][col] = accI[cc][v];
        }
    }
    __syncthreads();

    // ---- Phase 3: all three butterfly stages in registers, fused with scaled store ----
    // Thread task (r, k-pair): holds row r elements {j*64 + k, j*64 + k+1 : j=0..7}.
    // Stage s combines register slots at j-distance 2^s with twiddle tw_s[(j%2^s)*64 + k].
    for (int t = tid; t < ROWS * 32; t += NT) {
        const int r = t >> 5;
        const int k = (t & 31) << 1;          // even k; lanes hold consecutive pairs
        float er[16], ei[16];                 // slot(j,u) = 2*j + u, u = k offset 0/1
#pragma unroll
        for (int j = 0; j < 8; ++j) {
            er[2*j+0] = lr[r][j*64 + k];     er[2*j+1] = lr[r][j*64 + k + 1];
            ei[2*j+0] = li[r][j*64 + k];     ei[2*j+1] = li[r][j*64 + k + 1];
        }
#pragma unroll
        for (int u = 0; u < 2; ++u) {
            const int kk = k + u;
            // stage 0: pairs (2p, 2p+1), tw0[kk]
            const float w0r = twLr[kk], w0i = twLi[kk];
#pragma unroll
            for (int p = 0; p < 4; ++p)
                BFLY(4*p + u, 4*p + 2 + u, w0r, w0i);
            // stage 1: pairs (g*4 + j2, +2), tw1[j2*64 + kk]
#pragma unroll
            for (int g = 0; g < 2; ++g)
#pragma unroll
            for (int j2 = 0; j2 < 2; ++j2) {
                const float wr = twLr[64 + j2*64 + kk], wi_ = twLi[64 + j2*64 + kk];
                BFLY((g*4 + j2)*2 + u, (g*4 + j2 + 2)*2 + u, wr, wi_);
            }
            // stage 2: pairs (j2, j2+4), tw2[j2*64 + kk]
#pragma unroll
            for (int j2 = 0; j2 < 4; ++j2) {
                const float wr = twLr[192 + j2*64 + kk], wi_ = twLi[192 + j2*64 + kk];
                BFLY(j2*2 + u, (j2 + 4)*2 + u, wr, wi_);
            }
        }
        // scale by 2^-3 and store; lanes write consecutive float2 -> coalesced b64
        const long gbase = (rowBase + r) * NFFT + k;
#pragma unroll
        for (int j = 0; j < 8; ++j) {
            float2 sr, si;
            sr.x = er[2*j+0] * 0.125f;  sr.y = er[2*j+1] * 0.125f;
            si.x = ei[2*j+0] * 0.125f;  si.y = ei[2*j+1] * 0.125f;
            *reinterpret_cast<float2*>(outr + gbase + j*64) = sr;
            *reinterpret_cast<float2*>(outi + gbase + j*64) = si;
        }
    }
}

extern "C" void kernel_launch(void* const* d_in, const int* in_sizes, int n_in,
                              void* d_out, int out_size, void* d_ws, size_t ws_size,
                              hipStream_t stream)
{
    // setup_inputs order: re, im, M_re, M_im, tw_re0, tw_im0, tw_re1, tw_im1,
    //                     tw_re2, tw_im2, perm
    const float* re   = (const float*)d_in[0];
    const float* im   = (const float*)d_in[1];
    const float* Mre  = (const float*)d_in[2];
    const float* Mim  = (const float*)d_in[3];
    const float* tw0r = (const float*)d_in[4];
    const float* tw0i = (const float*)d_in[5];
    const float* tw1r = (const float*)d_in[6];
    const float* tw1i = (const float*)d_in[7];
    const float* tw2r = (const float*)d_in[8];
    const float* tw2i = (const float*)d_in[9];
    // d_in[10] = perm: its bit-reversal structure is folded into A addressing.

    float* outr = (float*)d_out;
    float* outi = outr + (size_t)in_sizes[0];

    const int rows   = in_sizes[0] / NFFT;   // 32*2000 = 64000
    const int blocks = rows / ROWS;          // 4000

    sparse_ifft_wmma<<<blocks, NT, 0, stream>>>(re, im, Mre, Mim,
                                                tw0r, tw0i, tw1r, tw1i,
                                                tw2r, tw2i, outr, outi);
    (void)d_ws; (void)ws_size; (void)n_in; (void)out_size;
}